// LightGCN_56453050138796
// MI455X (gfx1250) — compile-verified
//
#include <hip/hip_runtime.h>

#define EMBED_D 64
#define GNN_LAYERS 3

typedef __attribute__((ext_vector_type(2))) float v2f;
typedef __attribute__((ext_vector_type(8))) float v8f;

// pooled = cur = concat(user, item)
__global__ void lgcn_init(const float* __restrict__ user, const float* __restrict__ item,
                          float* __restrict__ cur, float* __restrict__ pooled,
                          int userElems, int totalElems) {
    int i = blockIdx.x * blockDim.x + threadIdx.x;
    if (i < totalElems) {
        float v = (i < userElems) ? user[i] : item[i - userElems];
        cur[i]    = v;
        pooled[i] = v;
    }
}

__global__ void lgcn_zero(float4* __restrict__ p, int n4) {
    int i = blockIdx.x * blockDim.x + threadIdx.x;
    if (i < n4) p[i] = make_float4(0.f, 0.f, 0.f, 0.f);
}

// One wave32 per edge: lane holds float2 of the 64-wide row.
// Gather cur[c] (256B coalesced), scatter v*x into next[r] via f32 atomics.
__global__ void lgcn_spmm_scatter(const int* __restrict__ rows, const int* __restrict__ cols,
                                  const float* __restrict__ vals, const float* __restrict__ cur,
                                  float* __restrict__ nxt, int E) {
    int wave = (blockIdx.x * blockDim.x + threadIdx.x) >> 5;
    int lane = threadIdx.x & 31;
    if (wave >= E) return;
    int   r = rows[wave];
    int   c = cols[wave];
    float v = vals[wave];
    const float2* src = (const float2*)(cur + (size_t)c * EMBED_D) + lane;
    float2 x = *src;
    float* dst = nxt + (size_t)r * EMBED_D + lane * 2;
    atomicAdd(dst,     v * x.x);
    atomicAdd(dst + 1, v * x.y);
}

// pooled += next, expressed as D = I16 x next_tile + pooled_tile via four chained
// V_WMMA_F32_16X16X4_F32 ops per 16x16 tile (identity is exact in f32).
// Layouts per ISA 7.12.2:
//   A (16x4): lane<16 -> M=lane, K={0,1}; lane>=16 -> M=lane-16, K={2,3}
//   B (4x16): VGPR j -> K = 2*half + j, N = lane%16
//   C/D (16x16): VGPR r -> M = r + 8*half, N = lane%16
__global__ void lgcn_pooled_acc_wmma(float* __restrict__ pooled, const float* __restrict__ nxt,
                                     int tiles) {
    int tile = (blockIdx.x * blockDim.x + threadIdx.x) >> 5;
    if (tile >= tiles) return;                 // wave-uniform branch: EXEC all-1s for WMMA
    int lane = threadIdx.x & 31;
    int half = lane >> 4;                      // 0 or 1
    int lc   = lane & 15;
    int m0 = (tile >> 2) << 4;                 // 4 column-tiles per row-block (D=64)
    int n0 = (tile & 3) << 4;

    const float* nb = nxt    + (size_t)m0 * EMBED_D + n0;
    float*       pb = pooled + (size_t)m0 * EMBED_D + n0;

    v8f c;
#pragma unroll
    for (int r = 0; r < 8; ++r)
        c[r] = pb[(size_t)(r + half * 8) * EMBED_D + lc];

#pragma unroll
    for (int k = 0; k < 4; ++k) {
        v2f a, b;
        a[0] = (lc == 4 * k + 2 * half + 0) ? 1.0f : 0.0f;
        a[1] = (lc == 4 * k + 2 * half + 1) ? 1.0f : 0.0f;
        b[0] = nb[(size_t)(4 * k + 2 * half + 0) * EMBED_D + lc];
        b[1] = nb[(size_t)(4 * k + 2 * half + 1) * EMBED_D + lc];
        // 8 args: (neg_a, A, neg_b, B, c_mod, C, reuse_a, reuse_b)
        c = __builtin_amdgcn_wmma_f32_16x16x4_f32(false, a, false, b, (short)0, c, false, false);
    }

#pragma unroll
    for (int r = 0; r < 8; ++r)
        pb[(size_t)(r + half * 8) * EMBED_D + lc] = c[r];
}

extern "C" void kernel_launch(void* const* d_in, const int* in_sizes, int n_in,
                              void* d_out, int out_size, void* d_ws, size_t ws_size,
                              hipStream_t stream) {
    const float* user = (const float*)d_in[0];
    const float* item = (const float*)d_in[1];
    const int*   rows = (const int*)d_in[2];
    const int*   cols = (const int*)d_in[3];
    const float* vals = (const float*)d_in[4];

    int userElems  = in_sizes[0];
    int totalElems = in_sizes[0] + in_sizes[1];   // N * 64
    int N = totalElems / EMBED_D;
    int E = in_sizes[2];

    float* pooled = (float*)d_out;
    float* bufA = (float*)d_ws;
    float* bufB = bufA + (size_t)totalElems;      // needs 2*N*D*4 = ~102 MB of ws

    lgcn_init<<<(totalElems + 255) / 256, 256, 0, stream>>>(user, item, bufA, pooled,
                                                            userElems, totalElems);

    float* cur = bufA;
    float* nxt = bufB;
    int tiles = (N / 16) * (EMBED_D / 16);        // 16x16 f32 tiles

    for (int l = 0; l < GNN_LAYERS; ++l) {
        lgcn_zero<<<((totalElems / 4) + 255) / 256, 256, 0, stream>>>((float4*)nxt,
                                                                      totalElems / 4);
        lgcn_spmm_scatter<<<(E + 7) / 8, 256, 0, stream>>>(rows, cols, vals, cur, nxt, E);
        lgcn_pooled_acc_wmma<<<(tiles + 7) / 8, 256, 0, stream>>>(pooled, nxt, tiles);
        float* t = cur; cur = nxt; nxt = t;
    }
}